// SGNN_58291296141978
// MI455X (gfx1250) — compile-verified
//
#include <hip/hip_runtime.h>
#include <math.h>

// Problem constants (fixed by the reference)
#define NN    64      // nodes
#define DM    128     // d_model
#define NIT   10      // NMF iterations
#define EPS   1e-8f
#define LD    66      // LDS row stride: even (8B-aligned b64 frag loads), 66%64=2 spreads banks
#define MSZ   (NN * LD)

typedef float v2f __attribute__((ext_vector_type(2)));
typedef float v8f __attribute__((ext_vector_type(8)));

__device__ __forceinline__ float softplusf(float x) {
    // numerically stable softplus
    return fmaxf(x, 0.0f) + log1pf(expf(-fabsf(x)));
}

// 64x64 = opA(Ma) @ opB(Mb), all in LDS with row stride LD (even).
// Wave `wave` owns a 32x32 block (2x2 tiles of 16x16):
//   rows [r0, r0+32), cols [c0, c0+32) with r0=(wave>>1)*32, c0=(wave&1)*32.
// Each k-step loads 2 A-frags + 2 B-frags and issues 4 WMMAs (1 load/WMMA).
// Contiguous fragment cases (non-transposed A, transposed B) use float2 (ds_load_b64).
template <bool TA, bool TB>
__device__ __forceinline__ void mm64(const float* Ma, const float* Mb, float* Mc,
                                     int wave, int lane) {
    const int half = lane >> 4;       // 0/1
    const int lq   = lane & 15;       // 0..15
    const int r0   = (wave >> 1) * 32;
    const int c0   = (wave & 1) * 32;
    v8f acc00 = {0.f,0.f,0.f,0.f,0.f,0.f,0.f,0.f};
    v8f acc01 = acc00, acc10 = acc00, acc11 = acc00;
    #pragma unroll
    for (int k0 = 0; k0 < NN; k0 += 4) {
        const int ka = k0 + 2 * half;   // K index of VGPR0 in this half (always even)
        v2f a0, a1, b0, b1;
        if (!TA) {
            a0 = *(const v2f*)&Ma[(r0 + lq) * LD + ka];        // contiguous -> b64
            a1 = *(const v2f*)&Ma[(r0 + 16 + lq) * LD + ka];
        } else {
            a0.x = Ma[(ka)     * LD + r0 + lq];
            a0.y = Ma[(ka + 1) * LD + r0 + lq];
            a1.x = Ma[(ka)     * LD + r0 + 16 + lq];
            a1.y = Ma[(ka + 1) * LD + r0 + 16 + lq];
        }
        if (!TB) {
            b0.x = Mb[(ka)     * LD + c0 + lq];
            b0.y = Mb[(ka + 1) * LD + c0 + lq];
            b1.x = Mb[(ka)     * LD + c0 + 16 + lq];
            b1.y = Mb[(ka + 1) * LD + c0 + 16 + lq];
        } else {
            b0 = *(const v2f*)&Mb[(c0 + lq) * LD + ka];        // contiguous -> b64
            b1 = *(const v2f*)&Mb[(c0 + 16 + lq) * LD + ka];
        }
        acc00 = __builtin_amdgcn_wmma_f32_16x16x4_f32(false, a0, false, b0, (short)0, acc00, false, false);
        acc01 = __builtin_amdgcn_wmma_f32_16x16x4_f32(false, a0, false, b1, (short)0, acc01, false, false);
        acc10 = __builtin_amdgcn_wmma_f32_16x16x4_f32(false, a1, false, b0, (short)0, acc10, false, false);
        acc11 = __builtin_amdgcn_wmma_f32_16x16x4_f32(false, a1, false, b1, (short)0, acc11, false, false);
    }
    #pragma unroll
    for (int v = 0; v < 8; ++v) {
        const int rA = (r0 + v + 8 * half) * LD;
        const int rB = (r0 + 16 + v + 8 * half) * LD;
        Mc[rA + c0 + lq]      = acc00[v];
        Mc[rA + c0 + 16 + lq] = acc01[v];
        Mc[rB + c0 + lq]      = acc10[v];
        Mc[rB + c0 + 16 + lq] = acc11[v];
    }
}

__global__ void __launch_bounds__(128)
sgnn_kernel(const float* __restrict__ Ag,   // (B,64,64)
            const float* __restrict__ Xg,   // (B,128,64)
            const float* __restrict__ xnew, // (B,128)
            const float* __restrict__ U0,   // (B,64,64)
            const float* __restrict__ V0,   // (B,64,64)
            float* __restrict__ out, int nB) {
    extern __shared__ float smem[];
    float* sA  = smem;
    float* sU  = smem + 1 * MSZ;
    float* sV  = smem + 2 * MSZ;
    float* sS1 = smem + 3 * MSZ;
    float* sS2 = smem + 4 * MSZ;
    float* sS3 = smem + 5 * MSZ;
    float* ex  = smem + 6 * MSZ;   // 512 floats of small vectors
    float* zb  = ex;               // [64]
    float* xpb = ex + 64;          // [128]
    float* tkb = ex + 192;         // [64]
    float* vsb = ex + 256;         // [64]
    float* rb  = ex + 320;         // [64]
    float* mxb = ex + 384;         // [64]
    float* ivb = ex + 448;         // [64]

    const int b    = blockIdx.x;
    const int tid  = threadIdx.x;
    const int wave = tid >> 5;
    const int lane = tid & 31;

    const float* gA = Ag + (size_t)b * (NN * NN);
    const float* gX = Xg + (size_t)b * (DM * NN);
    const float* gx = xnew + (size_t)b * DM;
    const float* gU = U0 + (size_t)b * (NN * NN);
    const float* gV = V0 + (size_t)b * (NN * NN);

    // ---- stage inputs into LDS (softplus on U0/V0) ----
    #pragma unroll 1
    for (int e = tid; e < NN * NN; e += 128) {
        const int idx = (e >> 6) * LD + (e & 63);
        sA[idx] = gA[e];
        sU[idx] = softplusf(gU[e]);
        sV[idx] = softplusf(gV[e]);
    }
    __syncthreads();

    // ---- NMF multiplicative updates ----
    #pragma unroll 1
    for (int it = 0; it < NIT; ++it) {
        // S1 = U^T A ; S2 = U^T U
        mm64<true, false>(sU, sA, sS1, wave, lane);
        mm64<true, false>(sU, sU, sS2, wave, lane);
        __syncthreads();
        // S3 = (U^T U) V
        mm64<false, false>(sS2, sV, sS3, wave, lane);
        __syncthreads();
        #pragma unroll 1
        for (int e = tid; e < NN * NN; e += 128) {
            const int idx = (e >> 6) * LD + (e & 63);
            sV[idx] = sV[idx] * sS1[idx] / (sS3[idx] + EPS);
        }
        __syncthreads();
        // S1 = A V^T ; S2 = U V
        mm64<false, true>(sA, sV, sS1, wave, lane);
        mm64<false, false>(sU, sV, sS2, wave, lane);
        __syncthreads();
        // S3 = (U V) V^T
        mm64<false, true>(sS2, sV, sS3, wave, lane);
        __syncthreads();
        #pragma unroll 1
        for (int e = tid; e < NN * NN; e += 128) {
            const int idx = (e >> 6) * LD + (e & 63);
            sU[idx] = sU[idx] * sS1[idx] / (sS3[idx] + EPS);
        }
        __syncthreads();
    }

    // ---- output pointers ----
    const size_t offXP = (size_t)nB * (NN * NN);
    const size_t offU  = offXP + (size_t)nB * DM;
    const size_t offV  = offU + (size_t)nB * (NN * NN);
    float* oA  = out + (size_t)b * (NN * NN);
    float* oXP = out + offXP + (size_t)b * DM;
    float* oU  = out + offU + (size_t)b * (NN * NN);
    float* oV  = out + offV + (size_t)b * (NN * NN);

    // ---- write U, V outputs ----
    #pragma unroll 1
    for (int e = tid; e < NN * NN; e += 128) {
        const int idx = (e >> 6) * LD + (e & 63);
        oU[e] = sU[idx];
        oV[e] = sV[idx];
    }

    // ---- z[n] = sum_d x_new[d] * X[d][n] ----
    if (tid < NN) {
        float acc = 0.f;
        #pragma unroll 1
        for (int d = 0; d < DM; ++d) acc += gx[d] * gX[d * NN + tid];
        zb[tid] = acc;
    }
    __syncthreads();

    // ---- x_p[d] = sum_n z[n] * X[d][n] ----
    {
        float acc = 0.f;
        #pragma unroll 1
        for (int n = 0; n < NN; ++n) acc += zb[n] * gX[tid * NN + n];
        xpb[tid] = acc;
        oXP[tid] = acc;
    }
    __syncthreads();

    // ---- tk[k] = (sum_d x_p[d] X[d][k])^2 / sqrt(D) ; vs[j] = rowsum(V)[j] ----
    if (tid < NN) {
        float s = 0.f;
        #pragma unroll 1
        for (int d = 0; d < DM; ++d) s += xpb[d] * gX[d * NN + tid];
        tkb[tid] = s * s * 0.08838834764831845f;   // 1/sqrt(128)
    } else {
        const int j = tid - NN;
        float acc = 0.f;
        #pragma unroll 1
        for (int k = 0; k < NN; ++k) acc += sV[j * LD + k];
        vsb[j] = acc;
    }
    __syncthreads();

    // ---- r[i] = sum_j U[i][j] vs[j]; per-row softmax stats of r[i]*tk[k] ----
    if (tid < NN) {
        float r = 0.f;
        #pragma unroll 1
        for (int j = 0; j < NN; ++j) r += sU[tid * LD + j] * vsb[j];
        float mx = -3.402823466e+38f;
        #pragma unroll 1
        for (int k = 0; k < NN; ++k) mx = fmaxf(mx, r * tkb[k]);
        float sum = 0.f;
        #pragma unroll 1
        for (int k = 0; k < NN; ++k) sum += expf(r * tkb[k] - mx);
        rb[tid]  = r;
        mxb[tid] = mx;
        ivb[tid] = 1.0f / sum;
    }
    __syncthreads();

    // ---- A_Tn (coalesced) ----
    #pragma unroll 1
    for (int e = tid; e < NN * NN; e += 128) {
        const int i = e >> 6, k = e & 63;
        oA[e] = expf(rb[i] * tkb[k] - mxb[i]) * ivb[i];
    }
}

extern "C" void kernel_launch(void* const* d_in, const int* in_sizes, int n_in,
                              void* d_out, int out_size, void* d_ws, size_t ws_size,
                              hipStream_t stream) {
    const float* Ag   = (const float*)d_in[0];
    const float* Xg   = (const float*)d_in[1];
    const float* xnew = (const float*)d_in[2];
    const float* U0   = (const float*)d_in[3];
    const float* V0   = (const float*)d_in[4];
    float* out = (float*)d_out;

    const int nB = in_sizes[2] / DM;   // x_new has B*D elements
    const size_t shmem = (size_t)(6 * MSZ + 512) * sizeof(float);
    sgnn_kernel<<<nB, 128, shmem, stream>>>(Ag, Xg, xnew, U0, V0, out, nB);
}